// SLoss_52570399703230
// MI455X (gfx1250) — compile-verified
//
#include <hip/hip_runtime.h>

// CDNA5 / gfx1250: wave32, V_WMMA_F32_16X16X4_F32 (D = A(16x4) * B(4x16) + C)
// Trick: pad K=3 -> 4 with A_k3 = 1.0 and B_k3 = yy, and pre-scale B by -2,
// so the WMMA directly yields  c = yy - 2*(x . y)  per (est_row, gt_col).
typedef __attribute__((ext_vector_type(2))) float v2f;
typedef __attribute__((ext_vector_type(8))) float v8f;

constexpr int BS = 16;
constexpr int D  = 3;
constexpr int N  = 4096;
constexpr int ROWS_PER_BLOCK = 128;             // 8 waves * 16 rows each
constexpr int JTILES = N / 16;                  // 256 gt tiles
constexpr int NBLK = BS * (N / ROWS_PER_BLOCK); // 512 workgroups
constexpr int PAD  = 16;                        // 16 float2 = 32 dwords: bank-split pad

// Phase 1: each workgroup handles (batch b, 128 est rows); writes one partial sum.
__global__ __launch_bounds__(256) void chamfer_phase1(
    const float* __restrict__ est,   // (BS, 3, N)
    const float* __restrict__ gt,    // (BS, 3, N)
    float* __restrict__ partial)     // NBLK floats
{
  // lo[k] = (-2*y0, -2*y1)  -> B rows K0,K1 (consumed by lanes 0-15)
  // hi[k] = (-2*y2,  yy  )  -> B rows K2,K3 (consumed by lanes 16-31)
  // PAD shifts hi by 32 dwords so the two half-waves hit disjoint bank groups.
  __shared__ v2f sh2[2 * N + PAD];

  const int tid     = threadIdx.x;
  const int lane    = tid & 31;
  const int wave    = tid >> 5;
  const int b       = blockIdx.x >> 5;          // / (N / ROWS_PER_BLOCK)
  const int iblk    = blockIdx.x & 31;
  const int lm      = lane & 15;
  const int halfsel = lane >> 4;                // 0: lanes 0-15, 1: lanes 16-31

  const float* est_b = est + (size_t)b * D * N;
  const float* gt_b  = gt  + (size_t)b * D * N;

  v2f* shlo = sh2;
  v2f* shhi = sh2 + N + PAD;

  // Cooperative staging: coalesced dword loads, pre-scaled by -2, yy folded in.
  for (int k = tid; k < N; k += 256) {
    float y0 = gt_b[k];
    float y1 = gt_b[N + k];
    float y2 = gt_b[2 * N + k];
    v2f lo, hi;
    lo[0] = -2.0f * y0;
    lo[1] = -2.0f * y1;
    hi[0] = -2.0f * y2;
    hi[1] = y0 * y0 + y1 * y1 + y2 * y2;        // yy
    shlo[k] = lo;
    shhi[k] = hi;
  }
  __syncthreads();

  // A fragment (16x4 f32): lanes 0-15: VGPR0=K0(x0), VGPR1=K1(x1);
  //                        lanes 16-31: VGPR0=K2(x2), VGPR1=K3(=1.0 -> picks up yy).
  const int i_base = iblk * ROWS_PER_BLOCK + wave * 16;
  v2f a;
  a[0] = halfsel ? est_b[2 * N + i_base + lm] : est_b[i_base + lm];
  a[1] = halfsel ? 1.0f                       : est_b[N + i_base + lm];

  // Per-lane B base pointer: selected once, no per-tile swizzle.
  const v2f* pb = (halfsel ? shhi : shlo) + lm;

  float mn[8];
#pragma unroll
  for (int v = 0; v < 8; ++v) mn[v] = 3.0e38f;

#pragma unroll 4
  for (int jt = 0; jt < JTILES; ++jt) {
    v2f bf = pb[jt * 16];                       // one ds_load_b64 per lane
    v8f c = {};
    // 8 args: (neg_a, A, neg_b, B, c_mod, C, reuse_a, reuse_b)
    c = __builtin_amdgcn_wmma_f32_16x16x4_f32(
        false, a, false, bf, (short)0, c, false, false);
    // c[v] = yy[Nidx] - 2*zz[M][Nidx];  M = i_base + v + halfsel*8, Nidx = jt*16 + lm
#pragma unroll
    for (int v = 0; v < 8; ++v) mn[v] = fminf(mn[v], c[v]);
  }

  // Min-reduce across the 16 lanes of each half (masks 1,2,4,8 stay in-half).
#pragma unroll
  for (int v = 0; v < 8; ++v) {
    for (int mask = 1; mask < 16; mask <<= 1)
      mn[v] = fminf(mn[v], __shfl_xor(mn[v], mask, 32));
  }

  __syncthreads();  // all waves done reading sh2; reuse its head for partials
  float* shp = reinterpret_cast<float*>(sh2);

  if (lm == 0) {  // lanes 0 and 16 hold the per-row minima
    float acc = 0.0f;
    const int rbase = i_base + halfsel * 8;
#pragma unroll
    for (int v = 0; v < 8; ++v) {
      const int r = rbase + v;
      float x0 = est_b[r];
      float x1 = est_b[N + r];
      float x2 = est_b[2 * N + r];
      acc += mn[v] + x0 * x0 + x1 * x1 + x2 * x2;   // add xx (constant in the min)
    }
    shp[wave * 2 + halfsel] = acc;
  }
  __syncthreads();

  if (tid == 0) {
    float s = 0.0f;
#pragma unroll
    for (int k = 0; k < 16; ++k) s += shp[k];
    partial[blockIdx.x] = s;
  }
}

// Phase 2: deterministic fixed-order reduction of the 512 partials.
__global__ __launch_bounds__(256) void chamfer_phase2(
    const float* __restrict__ partial, float* __restrict__ out)
{
  __shared__ float s[256];
  const int t = threadIdx.x;
  s[t] = partial[t] + partial[t + 256];
  __syncthreads();
  for (int off = 128; off > 0; off >>= 1) {
    if (t < off) s[t] += s[t + off];
    __syncthreads();
  }
  if (t == 0) out[0] = s[0] * (1.0f / (2.0f * (float)N * (float)BS));
}

extern "C" void kernel_launch(void* const* d_in, const int* in_sizes, int n_in,
                              void* d_out, int out_size, void* d_ws, size_t ws_size,
                              hipStream_t stream) {
  const float* est = (const float*)d_in[0];   // pts_est (16,3,4096) f32
  const float* gt  = (const float*)d_in[1];   // pts_gt  (16,3,4096) f32
  float* ws  = (float*)d_ws;                  // NBLK partial sums
  float* out = (float*)d_out;                 // scalar

  chamfer_phase1<<<NBLK, 256, 0, stream>>>(est, gt, ws);
  chamfer_phase2<<<1, 256, 0, stream>>>(ws, out);
}